// MemoryUnit_71399536329100
// MI455X (gfx1250) — compile-verified
//
#include <hip/hip_runtime.h>
#include <math.h>

// Problem constants (from reference)
#define DDIM  32768     // flattened feature dim = 512*8*8
#define BATCH 256       // batch
#define MSZ   2000      // memory slots
#define LAMV  0.0005f   // hard-shrink threshold
#define EPSN  1e-12f    // normalize eps

typedef __attribute__((ext_vector_type(2))) float v2f;
typedef __attribute__((ext_vector_type(8))) float v8f;

#define WMMA_F32(A, B, C) \
  __builtin_amdgcn_wmma_f32_16x16x4_f32(false, (A), false, (B), (short)0, (C), false, false)

// ---------------------------------------------------------------------------
// Kernel 1: per-row inverse L2 norms for feature (256 rows) and memory (2000
// rows), all rows length DDIM. One block per row.
// ---------------------------------------------------------------------------
__global__ __launch_bounds__(256) void norms_kernel(
    const float* __restrict__ feature, const float* __restrict__ memory,
    float* __restrict__ inv_f, float* __restrict__ inv_m) {
  const int row = blockIdx.x;
  const float* src;
  float* dst;
  int r;
  if (row < BATCH) { src = feature + (size_t)row * DDIM;           dst = inv_f; r = row; }
  else             { src = memory  + (size_t)(row - BATCH) * DDIM; dst = inv_m; r = row - BATCH; }

  float s = 0.f;
  for (int i = threadIdx.x * 4; i < DDIM; i += blockDim.x * 4) {
    float4 v = *(const float4*)(src + i);
    s += v.x * v.x + v.y * v.y + v.z * v.z + v.w * v.w;
  }
  __shared__ float red[256];
  red[threadIdx.x] = s;
  __syncthreads();
  for (int off = 128; off > 0; off >>= 1) {
    if (threadIdx.x < off) red[threadIdx.x] += red[threadIdx.x + off];
    __syncthreads();
  }
  if (threadIdx.x == 0) {
    float n = sqrtf(red[0]);
    dst[r] = 1.0f / fmaxf(n, EPSN);
  }
}

// ---------------------------------------------------------------------------
// Kernel 2: cosine-similarity GEMM
//   wraw[b,m] = (F[b,:] . M[m,:]) * inv_f[b] * inv_m[m]
// WMMA f32 16x16x4, fragments loaded directly from global as per-lane
// contiguous float2 (both A and B are K-pairs along row-major rows).
// Block: 256 threads = 8 waves; block tile 128(M) x 64(N).
// Wave tile: 32x32 = 2x2 WMMA tiles -> 4 wmma per 4 fragment loads.
// Ping-pong double-buffered K loop (step 8): two fragment sets alternate so
// each WMMA group waits on loads issued a full group earlier, with no
// register-rotation copies. kdim is a runtime arg to prevent full unroll.
// ---------------------------------------------------------------------------
__global__ __launch_bounds__(256) void gemm1_cos(
    const float* __restrict__ feature, const float* __restrict__ memory,
    const float* __restrict__ inv_f, const float* __restrict__ inv_m,
    float* __restrict__ wraw, int kdim) {
  const int lane = threadIdx.x & 31;
  const int wave = threadIdx.x >> 5;
  const int m0 = blockIdx.y * 128 + (wave & 3) * 32;   // 4 m-pairs per block
  const int n0 = blockIdx.x * 64  + (wave >> 2) * 32;  // 2 n-pairs per block

  const int lr   = lane & 15;         // row/col within tile
  const int koff = (lane >> 4) * 2;   // K sub-offset per lane half

  const int ra0 = m0 + lr;                        // feature rows (< 256)
  const int ra1 = m0 + 16 + lr;
  const int c0  = min(n0 + lr, MSZ - 1);          // clamp OOB memory rows
  const int c1  = min(n0 + 16 + lr, MSZ - 1);

  const float* ap0 = feature + (size_t)ra0 * DDIM + koff;
  const float* ap1 = feature + (size_t)ra1 * DDIM + koff;
  const float* bp0 = memory  + (size_t)c0  * DDIM + koff;
  const float* bp1 = memory  + (size_t)c1  * DDIM + koff;

  v8f acc00 = {}, acc01 = {}, acc10 = {}, acc11 = {};

  // Prologue: set A holds K-step 0.
  v2f a0A = *(const v2f*)(ap0);
  v2f a1A = *(const v2f*)(ap1);
  v2f b0A = *(const v2f*)(bp0);
  v2f b1A = *(const v2f*)(bp1);
  v2f a0B, a1B, b0B, b1B;

  int k = 0;
#pragma unroll 1
  for (; k < kdim - 8; k += 8) {
    // Load set B (k+4) while set A's loads drain, then consume set A.
    a0B = *(const v2f*)(ap0 + k + 4);
    a1B = *(const v2f*)(ap1 + k + 4);
    b0B = *(const v2f*)(bp0 + k + 4);
    b1B = *(const v2f*)(bp1 + k + 4);
    acc00 = WMMA_F32(a0A, b0A, acc00);
    acc01 = WMMA_F32(a0A, b1A, acc01);
    acc10 = WMMA_F32(a1A, b0A, acc10);
    acc11 = WMMA_F32(a1A, b1A, acc11);
    // Load set A (k+8), consume set B.
    a0A = *(const v2f*)(ap0 + k + 8);
    a1A = *(const v2f*)(ap1 + k + 8);
    b0A = *(const v2f*)(bp0 + k + 8);
    b1A = *(const v2f*)(bp1 + k + 8);
    acc00 = WMMA_F32(a0B, b0B, acc00);
    acc01 = WMMA_F32(a0B, b1B, acc01);
    acc10 = WMMA_F32(a1B, b0B, acc10);
    acc11 = WMMA_F32(a1B, b1B, acc11);
  }
  // Epilogue: set A holds kdim-8; final set B holds kdim-4.
  a0B = *(const v2f*)(ap0 + k + 4);
  a1B = *(const v2f*)(ap1 + k + 4);
  b0B = *(const v2f*)(bp0 + k + 4);
  b1B = *(const v2f*)(bp1 + k + 4);
  acc00 = WMMA_F32(a0A, b0A, acc00);
  acc01 = WMMA_F32(a0A, b1A, acc01);
  acc10 = WMMA_F32(a1A, b0A, acc10);
  acc11 = WMMA_F32(a1A, b1A, acc11);
  acc00 = WMMA_F32(a0B, b0B, acc00);
  acc01 = WMMA_F32(a0B, b1B, acc01);
  acc10 = WMMA_F32(a1B, b0B, acc10);
  acc11 = WMMA_F32(a1B, b1B, acc11);

  // C layout: VGPR v -> row base + (lane>=16 ? 8 : 0) + v, col tile_n + (lane&15)
  const int rb0 = m0 + (lane >> 4) * 8;        // rows for acc0x
  const int rb1 = rb0 + 16;                    // rows for acc1x
  // 2000 % 16 == 0, so a 16-wide tile is either fully valid or fully OOB.
  if (n0 < MSZ) {
    const int c = n0 + lr;
    const float im = inv_m[c];
#pragma unroll
    for (int v = 0; v < 8; ++v) {
      wraw[(size_t)(rb0 + v) * MSZ + c] = acc00[v] * inv_f[rb0 + v] * im;
      wraw[(size_t)(rb1 + v) * MSZ + c] = acc10[v] * inv_f[rb1 + v] * im;
    }
  }
  if (n0 + 16 < MSZ) {
    const int c = n0 + 16 + lr;
    const float im = inv_m[c];
#pragma unroll
    for (int v = 0; v < 8; ++v) {
      wraw[(size_t)(rb0 + v) * MSZ + c] = acc01[v] * inv_f[rb0 + v] * im;
      wraw[(size_t)(rb1 + v) * MSZ + c] = acc11[v] * inv_f[rb1 + v] * im;
    }
  }
}

// ---------------------------------------------------------------------------
// Kernel 3: softmax -> hard-shrink -> softmax along each 2000-wide row.
// One block (256 threads) per row, 8 elements per thread, LDS reductions.
// ---------------------------------------------------------------------------
__global__ __launch_bounds__(256) void dual_softmax(
    const float* __restrict__ wraw, float* __restrict__ wout) {
  const int row = blockIdx.x;
  const int t = threadIdx.x;
  const float* x = wraw + (size_t)row * MSZ;
  __shared__ float red[256];

  float v[8];
  float mx = -INFINITY;
#pragma unroll
  for (int j = 0; j < 8; ++j) {
    const int i = t + j * 256;
    v[j] = (i < MSZ) ? x[i] : -INFINITY;
    mx = fmaxf(mx, v[j]);
  }
  red[t] = mx; __syncthreads();
  for (int off = 128; off > 0; off >>= 1) {
    if (t < off) red[t] = fmaxf(red[t], red[t + off]);
    __syncthreads();
  }
  mx = red[0]; __syncthreads();

  float s = 0.f;
#pragma unroll
  for (int j = 0; j < 8; ++j) { v[j] = expf(v[j] - mx); s += v[j]; }
  red[t] = s; __syncthreads();
  for (int off = 128; off > 0; off >>= 1) {
    if (t < off) red[t] += red[t + off];
    __syncthreads();
  }
  s = red[0]; __syncthreads();

  const float inv1 = 1.f / s;
#pragma unroll
  for (int j = 0; j < 8; ++j) {
    const int i = t + j * 256;
    float w1 = v[j] * inv1;            // softmax #1 (w1 >= 0)
    w1 = (w1 > LAMV) ? w1 : 0.f;       // hard shrink
    v[j] = (i < MSZ) ? w1 : -INFINITY; // OOB lanes excluded from softmax #2
  }

  mx = -INFINITY;
#pragma unroll
  for (int j = 0; j < 8; ++j) mx = fmaxf(mx, v[j]);
  red[t] = mx; __syncthreads();
  for (int off = 128; off > 0; off >>= 1) {
    if (t < off) red[t] = fmaxf(red[t], red[t + off]);
    __syncthreads();
  }
  mx = red[0]; __syncthreads();

  s = 0.f;
#pragma unroll
  for (int j = 0; j < 8; ++j) { v[j] = expf(v[j] - mx); s += v[j]; }
  red[t] = s; __syncthreads();
  for (int off = 128; off > 0; off >>= 1) {
    if (t < off) red[t] += red[t + off];
    __syncthreads();
  }
  s = red[0];

  const float inv2 = 1.f / s;
#pragma unroll
  for (int j = 0; j < 8; ++j) {
    const int i = t + j * 256;
    if (i < MSZ) wout[(size_t)row * MSZ + i] = v[j] * inv2;
  }
}

// ---------------------------------------------------------------------------
// Kernel 4: mem_feat = w2 @ memory.  C[256,32768], K = 2000 (runtime kdim).
// A = w2 (row-major, per-lane contiguous float2 K-pairs).
// B = memory row-major [k][n]; per-lane column pairs -> two stride-DDIM loads,
// each B fragment reused by 2 wmma (2x2 register blocking).
// Ping-pong double-buffered K loop like gemm1.
// Block: 256 threads = 8 waves; block tile 128(M) x 64(N); dims divide evenly.
// ---------------------------------------------------------------------------
__global__ __launch_bounds__(256) void gemm2_out(
    const float* __restrict__ w2, const float* __restrict__ memory,
    float* __restrict__ out, int kdim) {
  const int lane = threadIdx.x & 31;
  const int wave = threadIdx.x >> 5;
  const int m0 = blockIdx.y * 128 + (wave & 3) * 32;
  const int n0 = blockIdx.x * 64  + (wave >> 2) * 32;

  const int lr   = lane & 15;
  const int koff = (lane >> 4) * 2;

  const int ra0 = m0 + lr;
  const int ra1 = m0 + 16 + lr;
  const float* ap0 = w2 + (size_t)ra0 * MSZ + koff;
  const float* ap1 = w2 + (size_t)ra1 * MSZ + koff;
  const float* bp0 = memory + (size_t)koff * DDIM + (n0 + lr);
  const float* bp1 = memory + (size_t)koff * DDIM + (n0 + 16 + lr);

  v8f acc00 = {}, acc01 = {}, acc10 = {}, acc11 = {};

  // Prologue: set A holds K-step 0.
  v2f a0A = *(const v2f*)(ap0);
  v2f a1A = *(const v2f*)(ap1);
  v2f b0A, b1A, a0B, a1B, b0B, b1B;
  b0A.x = bp0[0];
  b0A.y = bp0[DDIM];
  b1A.x = bp1[0];
  b1A.y = bp1[DDIM];

  int k = 0;
#pragma unroll 1
  for (; k < kdim - 8; k += 8) {
    {
      const size_t o = (size_t)(k + 4) * DDIM;
      a0B = *(const v2f*)(ap0 + k + 4);
      a1B = *(const v2f*)(ap1 + k + 4);
      b0B.x = bp0[o];
      b0B.y = bp0[o + DDIM];
      b1B.x = bp1[o];
      b1B.y = bp1[o + DDIM];
    }
    acc00 = WMMA_F32(a0A, b0A, acc00);
    acc01 = WMMA_F32(a0A, b1A, acc01);
    acc10 = WMMA_F32(a1A, b0A, acc10);
    acc11 = WMMA_F32(a1A, b1A, acc11);
    {
      const size_t o = (size_t)(k + 8) * DDIM;
      a0A = *(const v2f*)(ap0 + k + 8);
      a1A = *(const v2f*)(ap1 + k + 8);
      b0A.x = bp0[o];
      b0A.y = bp0[o + DDIM];
      b1A.x = bp1[o];
      b1A.y = bp1[o + DDIM];
    }
    acc00 = WMMA_F32(a0B, b0B, acc00);
    acc01 = WMMA_F32(a0B, b1B, acc01);
    acc10 = WMMA_F32(a1B, b0B, acc10);
    acc11 = WMMA_F32(a1B, b1B, acc11);
  }
  // Epilogue: set A holds kdim-8; final set B holds kdim-4.
  {
    const size_t o = (size_t)(k + 4) * DDIM;
    a0B = *(const v2f*)(ap0 + k + 4);
    a1B = *(const v2f*)(ap1 + k + 4);
    b0B.x = bp0[o];
    b0B.y = bp0[o + DDIM];
    b1B.x = bp1[o];
    b1B.y = bp1[o + DDIM];
  }
  acc00 = WMMA_F32(a0A, b0A, acc00);
  acc01 = WMMA_F32(a0A, b1A, acc01);
  acc10 = WMMA_F32(a1A, b0A, acc10);
  acc11 = WMMA_F32(a1A, b1A, acc11);
  acc00 = WMMA_F32(a0B, b0B, acc00);
  acc01 = WMMA_F32(a0B, b1B, acc01);
  acc10 = WMMA_F32(a1B, b0B, acc10);
  acc11 = WMMA_F32(a1B, b1B, acc11);

  const int rb0 = m0 + (lane >> 4) * 8;
  const int rb1 = rb0 + 16;
  const int cA = n0 + lr;
  const int cB = n0 + 16 + lr;
#pragma unroll
  for (int v = 0; v < 8; ++v) {
    out[(size_t)(rb0 + v) * DDIM + cA] = acc00[v];
    out[(size_t)(rb0 + v) * DDIM + cB] = acc01[v];
    out[(size_t)(rb1 + v) * DDIM + cA] = acc10[v];
    out[(size_t)(rb1 + v) * DDIM + cB] = acc11[v];
  }
}

// ---------------------------------------------------------------------------
extern "C" void kernel_launch(void* const* d_in, const int* in_sizes, int n_in,
                              void* d_out, int out_size, void* d_ws, size_t ws_size,
                              hipStream_t stream) {
  (void)in_sizes; (void)n_in; (void)out_size; (void)ws_size;

  const float* feature = (const float*)d_in[0];  // [256, 32768]
  const float* memory  = (const float*)d_in[1];  // [2000, 32768]

  float* out      = (float*)d_out;
  float* mem_feat = out;                               // [256, 32768]
  float* wfinal   = out + (size_t)BATCH * DDIM;        // [256, 2000]

  float* inv_f = (float*)d_ws;       // 256 floats
  float* inv_m = inv_f + BATCH;      // 2000 floats

  // Raw cosine scores staged in the mem_feat region (overwritten by gemm2
  // after dual_softmax has consumed them; regions are disjoint from wfinal).
  float* wraw = mem_feat;            // [256, 2000] scratch

  norms_kernel<<<BATCH + MSZ, 256, 0, stream>>>(feature, memory, inv_f, inv_m);
  gemm1_cos<<<dim3((MSZ + 63) / 64, BATCH / 128), 256, 0, stream>>>(
      feature, memory, inv_f, inv_m, wraw, DDIM);
  dual_softmax<<<BATCH, 256, 0, stream>>>(wraw, wfinal);
  gemm2_out<<<dim3(DDIM / 64, BATCH / 128), 256, 0, stream>>>(
      wfinal, memory, mem_feat, MSZ);
}